// TianCNN_23201413333534
// MI455X (gfx1250) — compile-verified
//
#include <hip/hip_runtime.h>

// Problem constants
#define XD    50          // channels
#define SEQ   230         // sequence length
#define NBATCH 16384
#define KDIM  150         // 3 * XD
#define KPAD  152         // padded to multiple of 4 for WMMA f32 16x16x4
#define NOUT  1380        // Y_DIM * SEQ_LEN
#define NPAD  1392        // 87 * 16
#define NTILES 87
#define MTILES 1024       // NBATCH / 16
#define MBLOCKS 512       // 2 M-tiles per block

typedef __attribute__((ext_vector_type(2))) float v2f;
typedef __attribute__((ext_vector_type(8))) float v8f;

// ---------------------------------------------------------------------------
// Stage 1: depthwise conv (k=3/6/12) + max over positions, per (batch,ch,conv)
// Rolling register window: each LDS element read once per task.
// ---------------------------------------------------------------------------
__device__ __forceinline__ float ldsx(const float* __restrict__ sh, int p, int c) {
    return sh[p * XD + c];
}

template <int K>
__device__ float conv_max(const float* __restrict__ sh, int c,
                          const float* __restrict__ wg, float bias) {
    const int OUT = SEQ - K + 1;
    float x[K];
#pragma unroll
    for (int i = 0; i < K; ++i) x[i] = ldsx(sh, i, c);
    float m = -3.402823466e38f;
    int p = 0;
    // main chunks of K positions; prefetch index p+s+K stays <= SEQ-1
    for (; p + 2 * K <= SEQ; p += K) {
#pragma unroll
        for (int s = 0; s < K; ++s) {
            float acc = bias;
#pragma unroll
            for (int i = 0; i < K; ++i)
                acc = fmaf(x[(s + i) % K], wg[i], acc);
            m = fmaxf(m, acc);
            x[s] = ldsx(sh, p + s + K, c);   // refill slot for next window
        }
    }
    // tail positions with direct loads
    for (; p < OUT; ++p) {
        float acc = bias;
#pragma unroll
        for (int i = 0; i < K; ++i)
            acc = fmaf(ldsx(sh, p + i, c), wg[i], acc);
        m = fmaxf(m, acc);
    }
    return m;
}

__global__ __launch_bounds__(160) void conv_max_kernel(
    const float* __restrict__ seq,
    const float* __restrict__ w1, const float* __restrict__ b1,
    const float* __restrict__ w2, const float* __restrict__ b2,
    const float* __restrict__ w3, const float* __restrict__ b3,
    float* __restrict__ flatp)
{
    __shared__ float sh[SEQ * XD];           // 46 KB of 320 KB/WGP
    const int b = blockIdx.x;
    {
        const float4* src = (const float4*)(seq + (size_t)b * (SEQ * XD));
        float4* dst = (float4*)sh;
        for (int i = threadIdx.x; i < (SEQ * XD) / 4; i += 160) dst[i] = src[i];
    }
    __syncthreads();

    const int t = threadIdx.x;
    if (t >= KPAD) return;
    float r = 0.0f;                          // K-padding entries must be zero
    if (t < KDIM) {
        const int c = t / 3;
        const int j = t - 3 * c;             // interleave order: (k3,k6,k12) per ch
        if (j == 0) {
            float w[3];
#pragma unroll
            for (int i = 0; i < 3; ++i) w[i] = w1[c * 3 + i];
            r = conv_max<3>(sh, c, w, b1[c]);
        } else if (j == 1) {
            float w[6];
#pragma unroll
            for (int i = 0; i < 6; ++i) w[i] = w2[c * 6 + i];
            r = conv_max<6>(sh, c, w, b2[c]);
        } else {
            float w[12];
#pragma unroll
            for (int i = 0; i < 12; ++i) w[i] = w3[c * 12 + i];
            r = conv_max<12>(sh, c, w, b3[c]);
        }
    }
    flatp[(size_t)b * KPAD + t] = r;
}

// ---------------------------------------------------------------------------
// Stage 2: pack w_out [1380,150] -> Wp2 pair-packed [(KPAD/2)][NPAD][2] so a
// lane's B fragment (two adjacent K rows, one column) is a single b64 load.
// Also pad bias to NPAD.
// ---------------------------------------------------------------------------
__global__ __launch_bounds__(256) void wpad_kernel(
    const float* __restrict__ w_out, const float* __restrict__ b_out,
    float* __restrict__ Wp2, float* __restrict__ Bp)
{
    const int idx = blockIdx.x * 256 + threadIdx.x;
    if (idx < KPAD * NPAD) {
        const int k = idx / NPAD;
        const int n = idx - k * NPAD;
        const float v = (k < KDIM && n < NOUT) ? w_out[(size_t)n * KDIM + k] : 0.0f;
        Wp2[(size_t)((k >> 1) * NPAD + n) * 2 + (k & 1)] = v;
    }
    if (idx < NPAD) Bp[idx] = (idx < NOUT) ? b_out[idx] : 0.0f;
}

// ---------------------------------------------------------------------------
// Stage 3: out[16384,1380] = flat @ W + bias, via V_WMMA_F32_16X16X4_F32.
// Block: 8 waves x 2 M-tiles each; per k-step one b64 B fragment feeds two
// WMMAs on independent accumulators (B reuse, no D->A/B hazard).
// ---------------------------------------------------------------------------
__global__ __launch_bounds__(256) void gemm_wmma_kernel(
    const float* __restrict__ flatp, const float* __restrict__ Wp2,
    const float* __restrict__ Bp, float* __restrict__ out)
{
    __shared__ float Ash[32 * KPAD];         // 19.4 KB
    const int mblk = blockIdx.x;             // 32 batch rows per block
    const int tid = threadIdx.x;
    {
        const float4* src = (const float4*)(flatp + (size_t)mblk * 32 * KPAD);
        float4* dst = (float4*)Ash;
        for (int i = tid; i < (32 * KPAD) / 4; i += 256) dst[i] = src[i];
    }
    __syncthreads();

    const int wave = tid >> 5;
    const int lane = tid & 31;
    const int ntile = blockIdx.y * 8 + wave;
    if (ntile >= NTILES) return;             // wave-uniform: EXEC stays all-1s

    const int half = lane >> 4;              // 0: K pair {0,1}, 1: K pair {2,3}
    const int l16  = lane & 15;
    const int ncol = ntile * 16 + l16;       // < NPAD always

    const float bias = Bp[ncol];
    v8f acc0 = {bias, bias, bias, bias, bias, bias, bias, bias};
    v8f acc1 = acc0;

    // A fragments: lane row = l16 (+16 for second M tile), K offset = 2*half
    const v2f* arow0 = (const v2f*)(Ash + l16 * KPAD);
    const v2f* arow1 = (const v2f*)(Ash + (16 + l16) * KPAD);
    // B fragments: pair-packed, one aligned b64 per k-step
    const v2f* brow = (const v2f*)Wp2 + (size_t)half * NPAD + ncol;

#pragma unroll 2
    for (int kk = 0; kk < KPAD / 4; ++kk) {
        v2f a0 = arow0[2 * kk + half];
        v2f a1 = arow1[2 * kk + half];
        v2f bf = brow[(size_t)(2 * kk) * NPAD];
        acc0 = __builtin_amdgcn_wmma_f32_16x16x4_f32(
            false, a0, false, bf, (short)0, acc0, false, false);
        acc1 = __builtin_amdgcn_wmma_f32_16x16x4_f32(
            false, a1, false, bf, (short)0, acc1, false, false);
    }

    // D layout: VGPR r -> row (half*8 + r), col = ncol
    if (ncol < NOUT) {
        float* op0 = out + (size_t)(mblk * 32 + half * 8) * NOUT + ncol;
        float* op1 = out + (size_t)(mblk * 32 + 16 + half * 8) * NOUT + ncol;
#pragma unroll
        for (int r = 0; r < 8; ++r) op0[(size_t)r * NOUT] = acc0[r];
#pragma unroll
        for (int r = 0; r < 8; ++r) op1[(size_t)r * NOUT] = acc1[r];
    }
}

// ---------------------------------------------------------------------------
extern "C" void kernel_launch(void* const* d_in, const int* in_sizes, int n_in,
                              void* d_out, int out_size, void* d_ws, size_t ws_size,
                              hipStream_t stream) {
    (void)in_sizes; (void)n_in; (void)out_size; (void)ws_size;
    const float* seq   = (const float*)d_in[0];
    const float* w1    = (const float*)d_in[1];
    const float* b1    = (const float*)d_in[2];
    const float* w2    = (const float*)d_in[3];
    const float* b2    = (const float*)d_in[4];
    const float* w3    = (const float*)d_in[5];
    const float* b3    = (const float*)d_in[6];
    const float* w_out = (const float*)d_in[7];
    const float* b_out = (const float*)d_in[8];
    float* outp = (float*)d_out;

    float* ws    = (float*)d_ws;
    float* flatp = ws;                                   // NBATCH*KPAD floats
    float* Wp2   = flatp + (size_t)NBATCH * KPAD;        // KPAD*NPAD floats
    float* Bp    = Wp2 + (size_t)KPAD * NPAD;            // NPAD floats

    conv_max_kernel<<<NBATCH, 160, 0, stream>>>(seq, w1, b1, w2, b2, w3, b3, flatp);

    const int wpad_elems = KPAD * NPAD;
    wpad_kernel<<<(wpad_elems + 255) / 256, 256, 0, stream>>>(w_out, b_out, Wp2, Bp);

    dim3 grid(MBLOCKS, (NTILES + 7) / 8);
    gemm_wmma_kernel<<<grid, 256, 0, stream>>>(flatp, Wp2, Bp, outp);
}